// CropRoi_50371376447837
// MI455X (gfx1250) — compile-verified
//
#include <hip/hip_runtime.h>
#include <hip/hip_bf16.h>

typedef __attribute__((ext_vector_type(2))) float v2f;
typedef __attribute__((ext_vector_type(8))) float v8f;
typedef __attribute__((ext_vector_type(4))) unsigned int u32x4;
typedef __attribute__((ext_vector_type(8))) unsigned int u32x8;

#define NPROP 64
#define EPSN  1e-5f

__device__ __forceinline__ int imin(int a, int b) { return a < b ? a : b; }

// pos = (p*11)/23 in f32 (matches jnp.arange(24)*(12-1)/(24-1))
__device__ __forceinline__ void up_idx(int p, int& i0, int& i1, float& w) {
    float pos = ((float)(p * 11)) / 23.0f;
    i0 = (int)pos;              // pos >= 0, truncation == floor
    w  = pos - (float)i0;
    i1 = imin(i0 + 1, 11);
}

__device__ __forceinline__ unsigned lds_offset(const void* p) {
    // generic pointer to LDS: addr[31:0] is the workgroup-relative LDS byte offset
    return (unsigned)(unsigned long long)(uintptr_t)p;
}

// ---------------------------------------------------------------------------
// TDM: DMA one crop z-slice (x=12, y=12, c=128) of comb2 into LDS.
// D# per CDNA5 ISA:
//   3D tensor, data_size=4B; addr = global + 4*(x + y*stride0 + c*stride1)
//   stride0 = 32 (row), stride1 = 32768 (channel); tile 12x12x128.
//   tensor dims = full 32x32x128 map -> no OOB. LDS layout: [c][y*12+x].
// ---------------------------------------------------------------------------
__device__ __forceinline__ void tdm_load_slice(const float* gaddr, unsigned ldsoff) {
    unsigned long long ga = (unsigned long long)(uintptr_t)gaddr;
    u32x4 g0;
    g0[0] = 1u;                                     // count=1 (valid), user mode
    g0[1] = ldsoff;                                 // lds_addr (bytes)
    g0[2] = (unsigned)(ga & 0xFFFFFFFFu);           // global_addr[31:0]
    g0[3] = (unsigned)((ga >> 32) & 0x01FFFFFFu)    // global_addr[56:32]
          | (2u << 30);                             // type = 2 ("image")
    u32x8 g1;
    g1[0] = 2u << 16;           // workgroup_mask=0 | data_size=2 (4 bytes)
    g1[1] = 32u << 16;          // tensor_dim0 = 32 (bits 48..63)
    g1[2] = 32u << 16;          // tensor_dim0_hi=0 | tensor_dim1 = 32 (bits 80..95)
    g1[3] = 12u << 16;          // tensor_dim1_hi=0 | tile_dim0 = 12 (x)
    g1[4] = 12u | (128u << 16); // tile_dim1 = 12 (y) | tile_dim2 = 128 (c)
    g1[5] = 32u;                // tensor_dim0_stride[31:0] = 32 (row stride)
    g1[6] = 32768u << 16;       // stride0_hi=0 | tensor_dim1_stride[15:0] (32768)
    g1[7] = 0u;                 // tensor_dim1_stride[47:16] = 0
    u32x4 g2;
    g2[0] = 128u;               // tensor_dim2 = 128 (channel extent)
    g2[1] = 0u;                 // tensor_dim3 (unused)
    g2[2] = 0u;                 // tensor_dim2_stride lo (unused: no 4th dim)
    g2[3] = 0u;                 // stride hi | tile_dim3 = 0 (unused)
    u32x4 g3;
    g3[0] = 0u; g3[1] = 0u; g3[2] = 0u; g3[3] = 0u;
    asm volatile("tensor_load_to_lds %0, %1, %2, %3"
                 :: "s"(g0), "s"(g1), "s"(g2), "s"(g3)
                 : "memory");
}

extern __shared__ float smem[];

// -------------------------------------------------------------------------
// Kernel 0: zero the stats2 accumulator
// -------------------------------------------------------------------------
__global__ void zeroK(float* p, int n) {
    int i = blockIdx.x * blockDim.x + threadIdx.x;
    if (i < n) p[i] = 0.0f;
}

// -------------------------------------------------------------------------
// Kernel A: conv1 at coarse resolution (commutes with trilinear upsample).
// g[p][64][1728] = w_up[64,128] x crop4[128,1728] + b_up
// One workgroup (8 waves) per proposal. TDM double-buffers one z-slice
// [128ch][144pts] of the crop into LDS while WMMA consumes the previous one.
// comb2: [2,128,32,32,32]  strides: b=4194304, c=32768, z=1024, y=32, x=1
// dyn LDS: wl[8192] bl[64] buf0[18432] buf1[18432]  = 180,480 B
// -------------------------------------------------------------------------
__global__ __launch_bounds__(256) void convA(const float* __restrict__ comb2,
                                             const int*   __restrict__ props,
                                             const float* __restrict__ w_up,
                                             const float* __restrict__ b_up,
                                             float*       __restrict__ g_ws) {
    float* wl   = smem;            // 8192
    float* bl   = wl + 8192;       // 64
    float* buf0 = bl + 64;         // 18432  [c][y*12+x]
    float* buf1 = buf0 + 18432;    // 18432

    const int p   = blockIdx.x;
    const int tid = threadIdx.x;
    for (int i = tid; i < 2048; i += 256)
        ((float4*)wl)[i] = ((const float4*)w_up)[i];
    if (tid < 64) bl[tid] = b_up[tid];

    const int* pr = props + p * 7;
    const int b  = __builtin_amdgcn_readfirstlane(pr[0]);
    const int z0 = __builtin_amdgcn_readfirstlane(pr[1]) >> 2;
    const int y0 = __builtin_amdgcn_readfirstlane(pr[2]) >> 2;
    const int x0 = __builtin_amdgcn_readfirstlane(pr[3]) >> 2;
    const float* src = comb2 + (size_t)b * 4194304 + (size_t)z0 * 1024 + y0 * 32 + x0;

    const int wave = tid >> 5, lane = tid & 31;
    const int n  = lane & 15;
    const int hi = (lane >= 16) ? 1 : 0;
    float* gout = g_ws + (size_t)p * 64 * 1728;

    __syncthreads();                       // weights staged
    if (wave == 0) tdm_load_slice(src, lds_offset(buf0));

    for (int z = 0; z < 12; ++z) {
        const float* buf = (z & 1) ? buf1 : buf0;
        if (wave == 0) {
            if (z + 1 < 12) {
                tdm_load_slice(src + (size_t)(z + 1) * 1024,
                               lds_offset((z & 1) ? buf0 : buf1));
                __builtin_amdgcn_s_wait_tensorcnt(1);   // slice z arrived
            } else {
                __builtin_amdgcn_s_wait_tensorcnt(0);
            }
        }
        __syncthreads();                   // hand-off: slice z visible to all waves

        for (int t = wave; t < 9; t += 8) {    // uniform per wave -> EXEC all-ones
            const int sl = t * 16 + n;         // point within slice [0,144)

            v8f acc[4];
            #pragma unroll
            for (int m = 0; m < 4; ++m)
                #pragma unroll
                for (int j = 0; j < 8; ++j)
                    acc[m][j] = bl[m * 16 + j + (hi ? 8 : 0)];

            for (int k = 0; k < 32; ++k) {
                const int c = k * 4 + (hi ? 2 : 0);
                v2f bf;
                bf.x = buf[c * 144 + sl];
                bf.y = buf[(c + 1) * 144 + sl];
                #pragma unroll
                for (int m = 0; m < 4; ++m) {
                    const int row = m * 16 + n;
                    v2f af;
                    af.x = wl[row * 128 + c];
                    af.y = wl[row * 128 + c + 1];
                    acc[m] = __builtin_amdgcn_wmma_f32_16x16x4_f32(
                        false, af, false, bf, (short)0, acc[m], false, false);
                }
            }
            #pragma unroll
            for (int m = 0; m < 4; ++m)
                #pragma unroll
                for (int j = 0; j < 8; ++j)
                    gout[(size_t)(m * 16 + j + (hi ? 8 : 0)) * 1728 + z * 144 + sl]
                        = acc[m][j];
        }
        __syncthreads();                   // all reads of buf done before overwrite
    }
}

// -------------------------------------------------------------------------
// Kernel B: InstanceNorm1 stats of up(g) via separable upsample in LDS.
// One workgroup per (proposal, channel). Emits folded (scale, shift).
// -------------------------------------------------------------------------
__global__ __launch_bounds__(256) void statsB(const float* __restrict__ g_ws,
                                              const float* __restrict__ g_up,
                                              const float* __restrict__ bt_up,
                                              float*       __restrict__ stats1) {
    __shared__ __align__(16) float gl[1728];  // [z12][y12][x12]
    __shared__ float ux[12 * 12 * 24];        // [z12][y12][x24]
    __shared__ float uxy[12 * 24 * 24];       // [z12][y24][x24]
    __shared__ float red[512];

    const int p = blockIdx.x >> 6, ch = blockIdx.x & 63;
    const int tid = threadIdx.x;
    const float* g = g_ws + ((size_t)p * 64 + ch) * 1728;

    for (int i = tid; i < 432; i += 256)
        ((float4*)gl)[i] = ((const float4*)g)[i];
    __syncthreads();

    for (int i = tid; i < 12 * 12 * 24; i += 256) {
        const int ox = i % 24, zy = i / 24;
        int i0, i1; float w; up_idx(ox, i0, i1, w);
        ux[i] = gl[zy * 12 + i0] * (1.0f - w) + gl[zy * 12 + i1] * w;
    }
    __syncthreads();
    for (int i = tid; i < 12 * 24 * 24; i += 256) {
        const int ox = i % 24, t = i / 24, oy = t % 24, z = t / 24;
        int i0, i1; float w; up_idx(oy, i0, i1, w);
        uxy[i] = ux[(z * 12 + i0) * 24 + ox] * (1.0f - w) +
                 ux[(z * 12 + i1) * 24 + ox] * w;
    }
    __syncthreads();
    float sum = 0.0f, ss = 0.0f;
    for (int i = tid; i < 13824; i += 256) {
        const int xy = i % 576, oz = i / 576;
        int i0, i1; float w; up_idx(oz, i0, i1, w);
        const float v = uxy[i0 * 576 + xy] * (1.0f - w) + uxy[i1 * 576 + xy] * w;
        sum += v; ss += v * v;
    }
    red[tid] = sum; red[256 + tid] = ss;
    __syncthreads();
    for (int st = 128; st > 0; st >>= 1) {
        if (tid < st) { red[tid] += red[tid + st]; red[256 + tid] += red[256 + tid + st]; }
        __syncthreads();
    }
    if (tid == 0) {
        const float mu  = red[0]   * (1.0f / 13824.0f);
        const float var = red[256] * (1.0f / 13824.0f) - mu * mu;
        const float rs  = rsqrtf(var + EPSN);
        const float sc  = g_up[ch] * rs;
        const float sh  = bt_up[ch] - mu * sc;
        stats1[(p * 64 + ch) * 2]     = sc;
        stats1[(p * 64 + ch) * 2 + 1] = sh;
    }
}

// -------------------------------------------------------------------------
// Kernel C: conv2 at fine resolution, one workgroup per (proposal, z-slice).
// B rows 0..63  = relu(norm(up(g))) built in LDS (point-major, pad 68 ->
//                 one ds_load_b64 per K-step, conflict-free banks).
// B rows 64..127 = crop2 streamed from out1 (L2 resident).
// Writes raw y2 and accumulates InstanceNorm2 sum/sumsq.
// out1: [2,64,64,64,64]  strides: b=16777216, c=262144, z=4096, y=64, x=1
// dyn LDS: 66,112 floats = 264,448 B
// -------------------------------------------------------------------------
#define UPAD 68
__global__ __launch_bounds__(256) void convC(const float* __restrict__ out1,
                                             const int*   __restrict__ props,
                                             const float* __restrict__ g_ws,
                                             const float* __restrict__ stats1,
                                             const float* __restrict__ w_b2,
                                             const float* __restrict__ b_b2,
                                             float*       __restrict__ y2_ws,
                                             float*       __restrict__ stats2) {
    float* wl    = smem;            // 8192   w_b2 [64][128]
    float* bl    = wl + 8192;       // 64
    float* scl   = bl + 64;         // 64
    float* shl   = scl + 64;        // 64
    float* gz    = shl + 64;        // 18432  [c64][2 z-planes][144]
    float* u     = gz + 18432;      // 39168  [pt576][UPAD] normalized+relu slab
    float* chsum = u + 576 * UPAD;  // 64
    float* chss  = chsum + 64;      // 64

    const int blk = blockIdx.x;
    const int p = blk / 24, zo = blk % 24;
    const int tid = threadIdx.x;

    for (int i = tid; i < 2048; i += 256)
        ((float4*)wl)[i] = ((const float4*)w_b2)[i];
    if (tid < 64) {
        bl[tid]  = b_b2[tid];
        scl[tid] = stats1[(p * 64 + tid) * 2];
        shl[tid] = stats1[(p * 64 + tid) * 2 + 1];
        chsum[tid] = 0.0f; chss[tid] = 0.0f;
    }

    int z_i0, z_i1; float wz; up_idx(zo, z_i0, z_i1, wz);

    const float* gp = g_ws + (size_t)p * 64 * 1728;
    for (int i = tid; i < 64 * 288; i += 256) {
        const int t = i % 288, c = i / 288;
        const int zz = t / 144, rem = t % 144;
        const int iz = zz ? z_i1 : z_i0;
        gz[i] = gp[(size_t)c * 1728 + iz * 144 + rem];
    }
    __syncthreads();

    for (int i = tid; i < 64 * 576; i += 256) {
        const int pt = i % 576, c = i / 576;
        const int oy = pt / 24, ox = pt % 24;
        int y0i, y1i, x0i, x1i; float wy, wx;
        up_idx(oy, y0i, y1i, wy);
        up_idx(ox, x0i, x1i, wx);
        const float* g0 = gz + c * 288;
        const float* g1 = g0 + 144;
        const float v00 = g0[y0i * 12 + x0i] * (1.0f - wx) + g0[y0i * 12 + x1i] * wx;
        const float v01 = g0[y1i * 12 + x0i] * (1.0f - wx) + g0[y1i * 12 + x1i] * wx;
        const float v10 = g1[y0i * 12 + x0i] * (1.0f - wx) + g1[y0i * 12 + x1i] * wx;
        const float v11 = g1[y1i * 12 + x0i] * (1.0f - wx) + g1[y1i * 12 + x1i] * wx;
        const float v0 = v00 * (1.0f - wy) + v01 * wy;
        const float v1 = v10 * (1.0f - wy) + v11 * wy;
        float v = (v0 * (1.0f - wz) + v1 * wz) * scl[c] + shl[c];
        u[pt * UPAD + c] = fmaxf(v, 0.0f);
    }
    __syncthreads();

    const int* pr = props + p * 7;
    const int b = pr[0], z0h = pr[1] >> 1, y0h = pr[2] >> 1, x0h = pr[3] >> 1;
    const float* c2base = out1 + (size_t)b * 16777216 +
                          (size_t)(z0h + zo) * 4096 + y0h * 64 + x0h;

    const int wave = tid >> 5, lane = tid & 31;
    const int n  = lane & 15;
    const int hi = (lane >= 16) ? 1 : 0;
    float* y2p = y2_ws + (size_t)p * 64 * 13824 + (size_t)zo * 576;

    for (int nt = wave; nt < 36; nt += 8) {   // uniform per wave
        const int pt = nt * 16 + n;
        const int yf = pt / 24, xf = pt % 24;
        const float* c2 = c2base + yf * 64 + xf;
        const float* urow = u + pt * UPAD;

        v8f acc[4];
        #pragma unroll
        for (int m = 0; m < 4; ++m)
            #pragma unroll
            for (int j = 0; j < 8; ++j)
                acc[m][j] = bl[m * 16 + j + (hi ? 8 : 0)];

        for (int k = 0; k < 32; ++k) {
            const int c = k * 4 + (hi ? 2 : 0);
            v2f bf;
            if (c < 64) {            // uniform branch: K-block never straddles 64
                bf = *(const v2f*)(urow + c);       // one ds_load_b64
            } else {
                bf.x = c2[(size_t)(c - 64) * 262144];
                bf.y = c2[(size_t)(c - 63) * 262144];
            }
            #pragma unroll
            for (int m = 0; m < 4; ++m) {
                const int row = m * 16 + n;
                v2f af;
                af.x = wl[row * 128 + c];
                af.y = wl[row * 128 + c + 1];
                acc[m] = __builtin_amdgcn_wmma_f32_16x16x4_f32(
                    false, af, false, bf, (short)0, acc[m], false, false);
            }
        }
        #pragma unroll
        for (int m = 0; m < 4; ++m) {
            #pragma unroll
            for (int j = 0; j < 8; ++j) {
                const int ch = m * 16 + j + (hi ? 8 : 0);
                const float v = acc[m][j];
                y2p[(size_t)ch * 13824 + pt] = v;
                float vs = v, vq = v * v;
                #pragma unroll
                for (int msk = 1; msk < 16; msk <<= 1) {
                    vs += __shfl_xor(vs, msk, 16);
                    vq += __shfl_xor(vq, msk, 16);
                }
                if (n == 0) { atomicAdd(&chsum[ch], vs); atomicAdd(&chss[ch], vq); }
            }
        }
    }
    __syncthreads();
    if (tid < 64) {
        atomicAdd(&stats2[(p * 64 + tid) * 2],     chsum[tid]);
        atomicAdd(&stats2[(p * 64 + tid) * 2 + 1], chss[tid]);
    }
}

// -------------------------------------------------------------------------
// Kernel D: InstanceNorm2 (affine per element) + ReLU + 4^3 max pool.
// One workgroup per (proposal, channel); 216 threads = 6x6x6 outputs.
// -------------------------------------------------------------------------
__global__ __launch_bounds__(216) void poolD(const float* __restrict__ y2_ws,
                                             const float* __restrict__ stats2,
                                             const float* __restrict__ g_b2,
                                             const float* __restrict__ bt_b2,
                                             float*       __restrict__ outp) {
    const int p = blockIdx.x >> 6, ch = blockIdx.x & 63;
    const float sum = stats2[(p * 64 + ch) * 2];
    const float ss  = stats2[(p * 64 + ch) * 2 + 1];
    const float mu  = sum * (1.0f / 13824.0f);
    const float var = ss * (1.0f / 13824.0f) - mu * mu;
    const float rs  = rsqrtf(var + EPSN);
    const float sc  = g_b2[ch] * rs;
    const float sh  = bt_b2[ch] - mu * sc;

    const int tid = threadIdx.x;
    const int oz = tid / 36, rr = tid % 36, oy = rr / 6, ox = rr % 6;
    const float* src = y2_ws + ((size_t)p * 64 + ch) * 13824;

    float mx = -INFINITY;
    #pragma unroll
    for (int dz = 0; dz < 4; ++dz)
        #pragma unroll
        for (int dy = 0; dy < 4; ++dy) {
            const float4 q = *(const float4*)(src + (oz * 4 + dz) * 576 +
                                              (oy * 4 + dy) * 24 + ox * 4);
            mx = fmaxf(mx, q.x * sc + sh);
            mx = fmaxf(mx, q.y * sc + sh);
            mx = fmaxf(mx, q.z * sc + sh);
            mx = fmaxf(mx, q.w * sc + sh);
        }
    outp[((size_t)(p * 64 + ch)) * 216 + tid] = fmaxf(mx, 0.0f);
}

// -------------------------------------------------------------------------
extern "C" void kernel_launch(void* const* d_in, const int* in_sizes, int n_in,
                              void* d_out, int out_size, void* d_ws, size_t ws_size,
                              hipStream_t stream) {
    // setup_inputs order: img, out1, comb2, inputs, proposals,
    //                     w_up, b_up, g_up, bt_up, w_b2, b_b2, g_b2, bt_b2
    const float* out1  = (const float*)d_in[1];
    const float* comb2 = (const float*)d_in[2];
    const int*   props = (const int*)  d_in[4];
    const float* w_up  = (const float*)d_in[5];
    const float* b_up  = (const float*)d_in[6];
    const float* g_up  = (const float*)d_in[7];
    const float* bt_up = (const float*)d_in[8];
    const float* w_b2  = (const float*)d_in[9];
    const float* b_b2  = (const float*)d_in[10];
    const float* g_b2  = (const float*)d_in[11];
    const float* bt_b2 = (const float*)d_in[12];
    float* outp = (float*)d_out;

    float* ws      = (float*)d_ws;
    float* g_ws    = ws;                                   // 64*64*1728
    float* stats1  = g_ws + (size_t)NPROP * 64 * 1728;     // 64*64*2
    float* stats2  = stats1 + (size_t)NPROP * 64 * 2;      // 64*64*2
    float* y2_ws   = stats2 + (size_t)NPROP * 64 * 2;      // 64*64*13824

    zeroK<<<(NPROP * 64 * 2 + 255) / 256, 256, 0, stream>>>(stats2, NPROP * 64 * 2);

    const size_t ldsA = (size_t)(8192 + 64 + 2 * 18432) * sizeof(float);
    convA<<<NPROP, 256, ldsA, stream>>>(comb2, props, w_up, b_up, g_ws);

    statsB<<<NPROP * 64, 256, 0, stream>>>(g_ws, g_up, bt_up, stats1);

    const size_t ldsC = (size_t)(8192 + 64 + 64 + 64 + 18432 + 576 * UPAD + 64 + 64)
                        * sizeof(float);
    convC<<<NPROP * 24, 256, ldsC, stream>>>(out1, props, g_ws, stats1,
                                             w_b2, b_b2, y2_ws, stats2);
    poolD<<<NPROP * 64, 216, 0, stream>>>(y2_ws, stats2, g_b2, bt_b2, outp);
}